// SimpleGCN_71536975282673
// MI455X (gfx1250) — compile-verified
//
#include <hip/hip_runtime.h>

typedef __attribute__((ext_vector_type(2))) float v2f;
typedef __attribute__((ext_vector_type(8))) float v8f;

// ---------------- init: deg=1 (self loop), zero group sums/counts ----------
__global__ void k_init(float* __restrict__ deg, float* __restrict__ gtail,
                       int N, int ntail) {
  int i = blockIdx.x * blockDim.x + threadIdx.x;
  if (i < N) deg[i] = 1.0f;
  if (i < ntail) gtail[i] = 0.0f;
}

// ---------------- in-degree over dst --------------------------------------
__global__ void k_deg(const int* __restrict__ dst, float* __restrict__ deg, int E) {
  int i = blockIdx.x * blockDim.x + threadIdx.x;
  if (i < E) atomicAdd(&deg[dst[i]], 1.0f);
}

// ---------------- dis = rsqrt(deg); cacc starts with self-loop term -------
__global__ void k_dis(const float* __restrict__ deg, float* __restrict__ dis,
                      float* __restrict__ cacc, int N) {
  int i = blockIdx.x * blockDim.x + threadIdx.x;
  if (i < N) {
    float r = rsqrtf(deg[i]);   // deg >= 1 always (self loops)
    dis[i] = r;
    cacc[i] = r;                // self loop contributes dis[src]==dis[i]
  }
}

// ---------------- cacc[d] += dis[src] over edges --------------------------
__global__ void k_cacc(const int* __restrict__ src, const int* __restrict__ dst,
                       const float* __restrict__ dis, float* __restrict__ cacc, int E) {
  int i = blockIdx.x * blockDim.x + threadIdx.x;
  if (i < E) atomicAdd(&cacc[dst[i]], dis[src[i]]);
}

// ---------------- tiny precompute: v1 = emb@W1, M = W2@fcW (pad 16),
//                  bout = b2@fcW + fcb.  One block of H(=256) threads. ------
__global__ void k_small(const float* __restrict__ emb, const float* __restrict__ W1,
                        const float* __restrict__ W2, const float* __restrict__ b2,
                        const float* __restrict__ fcW, const float* __restrict__ fcb,
                        float* __restrict__ v1, float* __restrict__ M,
                        float* __restrict__ bout, int D, int H, int O) {
  int j = threadIdx.x;
  float s = 0.f;
  for (int k = 0; k < D; ++k) s = fmaf(emb[k], W1[k * H + j], s);
  v1[j] = s;
  for (int o = 0; o < 16; ++o) {
    float m = 0.f;
    if (o < O)
      for (int l = 0; l < H; ++l) m = fmaf(W2[j * H + l], fcW[l * O + o], m);
    M[j * 16 + o] = m;
  }
  if (j < 16) {
    float bb = 0.f;
    if (j < O) {
      for (int l = 0; l < H; ++l) bb = fmaf(b2[l], fcW[l * O + j], bb);
      bb += fcb[j];
    }
    bout[j] = bb;
  }
}

// ---------------- WMMA node kernel: t[i] = relu(c1[i]*v1 + b1) @ M --------
// One wave handles a 16-node tile; fp32 WMMA 16x16x4 accumulates K=256.
// A (16x4 f32): v0 = K={0|2}, v1 = K={1|3} split by lane half.  B mirrored.
__global__ void __launch_bounds__(256) k_node_wmma(
    const float* __restrict__ dis, const float* __restrict__ cacc,
    const float* __restrict__ b1,  const float* __restrict__ v1g,
    const float* __restrict__ M,   float* __restrict__ t,
    int N, int ntiles) {
  __shared__ float sV[256];
  __shared__ float sB[256];
  __shared__ float sM[256 * 16];
  int tid = threadIdx.x;
  sV[tid] = v1g[tid];
  sB[tid] = b1[tid];
  for (int i = tid; i < 256 * 16; i += 256) sM[i] = M[i];
  __syncthreads();

  int wave = tid >> 5;
  int lane = tid & 31;
  int tile = blockIdx.x * 8 + wave;
  if (tile >= ntiles) return;   // wave-uniform: EXEC stays all-ones for WMMA

  int r   = lane & 15;          // row (A) / column (B, D)
  int hi  = lane >> 4;          // K sub-pair select
  int row = tile * 16 + r;
  int rowc = row < N ? row : N - 1;
  float c = dis[rowc] * cacc[rowc];   // c1 for this node

  v8f acc = {0.f, 0.f, 0.f, 0.f, 0.f, 0.f, 0.f, 0.f};
  #pragma unroll 4
  for (int kk = 0; kk < 64; ++kk) {
    int k = kk * 4 + hi * 2;
    float a0 = fmaxf(fmaf(c, sV[k],     sB[k]),     0.f);
    float a1 = fmaxf(fmaf(c, sV[k + 1], sB[k + 1]), 0.f);
    v2f a = {a0, a1};
    v2f b = {sM[k * 16 + r], sM[(k + 1) * 16 + r]};
    acc = __builtin_amdgcn_wmma_f32_16x16x4_f32(
        /*neg_a=*/false, a, /*neg_b=*/false, b,
        /*c_mod=*/(short)0, acc, /*reuse_a=*/false, /*reuse_b=*/false);
  }
  // D layout: lanes 0-15 col=lane rows 0..7 in v0..v7; lanes 16-31 rows 8..15
  if (r < 10) {
    #pragma unroll
    for (int v = 0; v < 8; ++v) {
      int orow = tile * 16 + hi * 8 + v;
      if (orow < N) t[(size_t)orow * 16 + r] = acc[v];
    }
  }
}

// ---------------- edge + self-loop aggregation into 64x10 group sums ------
__global__ void __launch_bounds__(256) k_agg(
    const int* __restrict__ src, const int* __restrict__ dst,
    const int* __restrict__ batch, const float* __restrict__ dis,
    const float* __restrict__ t, float* __restrict__ gsum,
    float* __restrict__ gcnt, int E, int N, int G) {
  __shared__ float ls[96 * 10];   // supports G up to 96
  __shared__ float lc[96];
  for (int i = threadIdx.x; i < G * 10; i += blockDim.x) ls[i] = 0.f;
  for (int i = threadIdx.x; i < G; i += blockDim.x) lc[i] = 0.f;
  __syncthreads();

  int T = E + N;
  int stride = gridDim.x * blockDim.x;
  for (int idx = blockIdx.x * blockDim.x + threadIdx.x; idx < T; idx += stride) {
    int s, d; float w;
    if (idx < E) { s = src[idx]; d = dst[idx]; w = dis[s] * dis[d]; }
    else         { s = idx - E;  d = s; float rr = dis[s]; w = rr * rr; }
    int g = batch[d];
    const float* tv = t + (size_t)s * 16;
    float* lg = ls + g * 10;
    #pragma unroll
    for (int o = 0; o < 10; ++o) atomicAdd(&lg[o], w * tv[o]);  // ds_add_f32
    if (idx >= E) atomicAdd(&lc[g], 1.0f);
  }
  __syncthreads();
  for (int i = threadIdx.x; i < G * 10; i += blockDim.x) atomicAdd(&gsum[i], ls[i]);
  for (int i = threadIdx.x; i < G; i += blockDim.x) atomicAdd(&gcnt[i], lc[i]);
}

// ---------------- finalize: divide by counts, add b2@fcW + fcb ------------
__global__ void k_final(const float* __restrict__ gsum, const float* __restrict__ gcnt,
                        const float* __restrict__ bout, const float* __restrict__ fcb,
                        float* __restrict__ out, int G, int O) {
  int i = blockIdx.x * blockDim.x + threadIdx.x;
  if (i < G * O) {
    int g = i / O, o = i % O;
    float cnt = gcnt[g];
    out[i] = (cnt > 0.f) ? (gsum[g * O + o] / cnt + bout[o]) : fcb[o];
  }
}

extern "C" void kernel_launch(void* const* d_in, const int* in_sizes, int n_in,
                              void* d_out, int out_size, void* d_ws, size_t ws_size,
                              hipStream_t stream) {
  (void)n_in; (void)ws_size;
  const int*   edge  = (const int*)d_in[1];
  const int*   batch = (const int*)d_in[2];
  const float* emb   = (const float*)d_in[3];
  const float* W1    = (const float*)d_in[4];
  const float* b1    = (const float*)d_in[5];
  const float* W2    = (const float*)d_in[6];
  const float* b2    = (const float*)d_in[7];
  const float* fcW   = (const float*)d_in[8];
  const float* fcb   = (const float*)d_in[9];

  int N = in_sizes[0];
  int E = in_sizes[1] / 2;
  int D = in_sizes[3];
  int H = in_sizes[5];     // 256
  int O = in_sizes[9];     // 10
  int G = out_size / O;    // 64
  const int* src = edge;
  const int* dst = edge + E;

  float* ws = (float*)d_ws;
  size_t off = 0;
  float* deg  = ws + off; off += (size_t)N;
  float* dis  = ws + off; off += (size_t)N;
  float* cacc = ws + off; off += (size_t)N;
  float* t    = ws + off; off += (size_t)N * 16;
  float* v1   = ws + off; off += (size_t)H;
  float* M    = ws + off; off += (size_t)H * 16;
  float* bout = ws + off; off += 16;
  float* gsum = ws + off; off += (size_t)G * 10;
  float* gcnt = ws + off; off += (size_t)G;

  const int B = 256;
  int ntail = G * 10 + G;                 // gsum + gcnt contiguous
  int ninit = N > ntail ? N : ntail;
  k_init<<<(ninit + B - 1) / B, B, 0, stream>>>(deg, gsum, N, ntail);
  k_deg<<<(E + B - 1) / B, B, 0, stream>>>(dst, deg, E);
  k_dis<<<(N + B - 1) / B, B, 0, stream>>>(deg, dis, cacc, N);
  k_cacc<<<(E + B - 1) / B, B, 0, stream>>>(src, dst, dis, cacc, E);
  k_small<<<1, H, 0, stream>>>(emb, W1, W2, b2, fcW, fcb, v1, M, bout, D, H, O);
  int ntiles = (N + 15) / 16;
  k_node_wmma<<<(ntiles + 7) / 8, 256, 0, stream>>>(dis, cacc, b1, v1, M, t, N, ntiles);
  k_agg<<<128, 256, 0, stream>>>(src, dst, batch, dis, t, gsum, gcnt, E, N, G);
  k_final<<<(G * O + B - 1) / B, B, 0, stream>>>(gsum, gcnt, bout, fcb,
                                                 (float*)d_out, G, O);
}